// SelfAttentionBlock_87376814669850
// MI455X (gfx1250) — compile-verified
//
#include <hip/hip_runtime.h>
#include <hip/hip_bf16.h>

// Problem constants (from reference)
#define BB 2
#define CC 64
#define CQ 8
#define NN 8192            // T*W*H = 8*32*32
#define NT (NN / 16)       // 512 row/col tiles of 16
#define EPS 1e-5f

typedef __attribute__((ext_vector_type(2))) float v2f;
typedef __attribute__((ext_vector_type(8))) float v8f;

// D = A(16x4) * B(4x16) + C, f32 everywhere.
// 8-arg form: (neg_a, A, neg_b, B, c_mod, C, reuse_a, reuse_b)
static __device__ __forceinline__ v8f wmma_f32_k4(v2f a, v2f b, v8f c) {
  return __builtin_amdgcn_wmma_f32_16x16x4_f32(false, a, false, b, (short)0, c,
                                               false, false);
}

// ---------------------------------------------------------------------------
// Kernel 1: q/k/v 1x1 conv projections.  One thread per (b, n) position.
// ---------------------------------------------------------------------------
__global__ void proj_kernel(const float* __restrict__ x,
                            const float* __restrict__ q_w, const float* __restrict__ q_b,
                            const float* __restrict__ k_w, const float* __restrict__ k_b,
                            const float* __restrict__ v_w, const float* __restrict__ v_b,
                            float* __restrict__ q, float* __restrict__ k,
                            float* __restrict__ v) {
  const int idx = blockIdx.x * blockDim.x + threadIdx.x;   // over B*N
  const int b = idx / NN;
  const int n = idx % NN;
  const float* xb = x + (size_t)b * CC * NN + n;

  float xv[CC];
#pragma unroll
  for (int c = 0; c < CC; ++c) xv[c] = xb[(size_t)c * NN];

#pragma unroll
  for (int o = 0; o < CQ; ++o) {
    float aq = q_b[o], ak = k_b[o];
#pragma unroll
    for (int c = 0; c < CC; ++c) {
      aq = fmaf(q_w[o * CC + c], xv[c], aq);
      ak = fmaf(k_w[o * CC + c], xv[c], ak);
    }
    q[((size_t)b * CQ + o) * NN + n] = aq;
    k[((size_t)b * CQ + o) * NN + n] = ak;
  }
  for (int o = 0; o < CC; ++o) {
    float av = v_b[o];
#pragma unroll
    for (int c = 0; c < CC; ++c) av = fmaf(v_w[o * CC + c], xv[c], av);
    v[((size_t)b * CC + o) * NN + n] = av;
  }
}

// ---------------------------------------------------------------------------
// Kernel 2: per-row softmax stats (max, sum of exp) via WMMA energy recompute.
// One wave per 16-row stripe of energy[b, :, :].  Energy never hits memory.
// ---------------------------------------------------------------------------
__global__ void attn_rowstats_kernel(const float* __restrict__ q,
                                     const float* __restrict__ kmat,
                                     float* __restrict__ rmax,
                                     float* __restrict__ rsum) {
  const int lane = threadIdx.x & 31;
  const int wave = threadIdx.x >> 5;
  const int g = blockIdx.x * (blockDim.x >> 5) + wave;  // tile id over B*NT
  const int b = g / NT;
  const int i0 = (g % NT) * 16;
  const int m = lane & 15;
  const int kh = (lane >> 4) * 2;        // K split across half-waves
  const int rbase = (lane >> 4) * 8;     // rows 0..7 vs 8..15 in D layout

  const float* qb = q + (size_t)b * CQ * NN;
  const float* kb = kmat + (size_t)b * CQ * NN;

  v2f a0, a1;                            // A = q^T stripe, 16 x 8 (two K=4 tiles)
  a0.x = qb[(size_t)(kh + 0) * NN + i0 + m];
  a0.y = qb[(size_t)(kh + 1) * NN + i0 + m];
  a1.x = qb[(size_t)(kh + 4) * NN + i0 + m];
  a1.y = qb[(size_t)(kh + 5) * NN + i0 + m];

  float lm[8], ls[8];
#pragma unroll
  for (int r = 0; r < 8; ++r) { lm[r] = -__builtin_inff(); ls[r] = 0.f; }

  for (int j0 = 0; j0 < NN; j0 += 16) {
    v2f b0, b1;
    b0.x = kb[(size_t)(kh + 0) * NN + j0 + m];
    b0.y = kb[(size_t)(kh + 1) * NN + j0 + m];
    b1.x = kb[(size_t)(kh + 4) * NN + j0 + m];
    b1.y = kb[(size_t)(kh + 5) * NN + j0 + m];
    v8f c = {0.f, 0.f, 0.f, 0.f, 0.f, 0.f, 0.f, 0.f};
    c = wmma_f32_k4(a0, b0, c);
    c = wmma_f32_k4(a1, b1, c);
#pragma unroll
    for (int r = 0; r < 8; ++r) {        // online softmax update per row slot
      float e = c[r];
      float mn = fmaxf(lm[r], e);
      ls[r] = ls[r] * __expf(lm[r] - mn) + __expf(e - mn);
      lm[r] = mn;
    }
  }
  // Merge the 16 columns held across each half-wave (masks stay within half).
#pragma unroll
  for (int mask = 1; mask <= 8; mask <<= 1) {
#pragma unroll
    for (int r = 0; r < 8; ++r) {
      float om = __shfl_xor(lm[r], mask, 32);
      float os = __shfl_xor(ls[r], mask, 32);
      float mm = fmaxf(lm[r], om);
      ls[r] = ls[r] * __expf(lm[r] - mm) + os * __expf(om - mm);
      lm[r] = mm;
    }
  }
  if ((lane & 15) == 0) {
#pragma unroll
    for (int r = 0; r < 8; ++r) {
      rmax[(size_t)b * NN + i0 + rbase + r] = lm[r];
      rsum[(size_t)b * NN + i0 + rbase + r] = ls[r];
    }
  }
}

// ---------------------------------------------------------------------------
// Kernel 3: recompute energy tiles, normalize, single NT write of attention.
// ---------------------------------------------------------------------------
__global__ void attn_write_kernel(const float* __restrict__ q,
                                  const float* __restrict__ kmat,
                                  const float* __restrict__ rmax,
                                  const float* __restrict__ rsum,
                                  float* __restrict__ attn) {
  const int lane = threadIdx.x & 31;
  const int wave = threadIdx.x >> 5;
  const int g = blockIdx.x * (blockDim.x >> 5) + wave;
  const int b = g / NT;
  const int i0 = (g % NT) * 16;
  const int m = lane & 15;
  const int kh = (lane >> 4) * 2;
  const int rbase = (lane >> 4) * 8;

  const float* qb = q + (size_t)b * CQ * NN;
  const float* kb = kmat + (size_t)b * CQ * NN;

  v2f a0, a1;
  a0.x = qb[(size_t)(kh + 0) * NN + i0 + m];
  a0.y = qb[(size_t)(kh + 1) * NN + i0 + m];
  a1.x = qb[(size_t)(kh + 4) * NN + i0 + m];
  a1.y = qb[(size_t)(kh + 5) * NN + i0 + m];

  float mrow[8], isrow[8];
#pragma unroll
  for (int r = 0; r < 8; ++r) {
    mrow[r] = rmax[(size_t)b * NN + i0 + rbase + r];
    isrow[r] = 1.f / rsum[(size_t)b * NN + i0 + rbase + r];  // sum >= 1
  }

  float* arow = attn + (size_t)b * NN * NN;
  for (int j0 = 0; j0 < NN; j0 += 16) {
    v2f b0, b1;
    b0.x = kb[(size_t)(kh + 0) * NN + j0 + m];
    b0.y = kb[(size_t)(kh + 1) * NN + j0 + m];
    b1.x = kb[(size_t)(kh + 4) * NN + j0 + m];
    b1.y = kb[(size_t)(kh + 5) * NN + j0 + m];
    v8f c = {0.f, 0.f, 0.f, 0.f, 0.f, 0.f, 0.f, 0.f};
    c = wmma_f32_k4(a0, b0, c);
    c = wmma_f32_k4(a1, b1, c);
#pragma unroll
    for (int r = 0; r < 8; ++r) {
      float av = __expf(c[r] - mrow[r]) * isrow[r];
      __builtin_nontemporal_store(
          av, arow + (size_t)(i0 + rbase + r) * NN + (j0 + m));
    }
  }
}

// ---------------------------------------------------------------------------
// Kernel 4: out[b,c,i] = sum_j v[b,c,j] * attn[b,i,j]  (WMMA, K=8192).
// One wave per 16x16 output tile; NT loads of attention (its only read).
// ---------------------------------------------------------------------------
__global__ void out_gemm_kernel(const float* __restrict__ v,
                                const float* __restrict__ attn,
                                float* __restrict__ outm) {
  const int lane = threadIdx.x & 31;
  const int wave = threadIdx.x >> 5;
  const int g = blockIdx.x * (blockDim.x >> 5) + wave;  // over B*(C/16)*NT
  const int b = g / ((CC / 16) * NT);
  const int rem = g % ((CC / 16) * NT);
  const int c0 = (rem / NT) * 16;
  const int i0 = (rem % NT) * 16;
  const int m = lane & 15;
  const int kh = (lane >> 4) * 2;
  const int rbase = (lane >> 4) * 8;

  const float* vb = v + (size_t)b * CC * NN;
  const float* ab = attn + (size_t)b * NN * NN;

  v8f acc = {0.f, 0.f, 0.f, 0.f, 0.f, 0.f, 0.f, 0.f};
  for (int j0 = 0; j0 < NN; j0 += 8) {
    // A = v rows c0..c0+15, K = j (contiguous pairs -> b64 loads)
    v2f a0 = *(const v2f*)(vb + (size_t)(c0 + m) * NN + j0 + kh);
    v2f a1 = *(const v2f*)(vb + (size_t)(c0 + m) * NN + j0 + 4 + kh);
    // B = attn^T: element (K=j, N=i) = attn[i0+m][j...], streamed NT
    v2f b0 = __builtin_nontemporal_load(
        (const v2f*)(ab + (size_t)(i0 + m) * NN + j0 + kh));
    v2f b1 = __builtin_nontemporal_load(
        (const v2f*)(ab + (size_t)(i0 + m) * NN + j0 + 4 + kh));
    acc = wmma_f32_k4(a0, b0, acc);
    acc = wmma_f32_k4(a1, b1, acc);
  }
#pragma unroll
  for (int r = 0; r < 8; ++r)
    outm[((size_t)b * CC + c0 + rbase + r) * NN + i0 + m] = acc[r];
}

// ---------------------------------------------------------------------------
// Kernel 5: W 1x1 conv.  One thread per (b, n).
// ---------------------------------------------------------------------------
__global__ void wproj_kernel(const float* __restrict__ outm,
                             const float* __restrict__ w_w,
                             const float* __restrict__ w_b,
                             float* __restrict__ wout) {
  const int idx = blockIdx.x * blockDim.x + threadIdx.x;  // over B*N
  const int b = idx / NN;
  const int n = idx % NN;
  const float* ob = outm + (size_t)b * CC * NN + n;

  float xv[CC];
#pragma unroll
  for (int c = 0; c < CC; ++c) xv[c] = ob[(size_t)c * NN];
  for (int o = 0; o < CC; ++o) {
    float a = w_b[o];
#pragma unroll
    for (int c = 0; c < CC; ++c) a = fmaf(w_w[o * CC + c], xv[c], a);
    wout[((size_t)b * CC + o) * NN + n] = a;
  }
}

// ---------------------------------------------------------------------------
// Kernel 6: per-channel batch stats (sum, sumsq) — one block per channel.
// ---------------------------------------------------------------------------
__global__ void stats_kernel(const float* __restrict__ wout,
                             float* __restrict__ stats) {
  const int o = blockIdx.x;
  float s = 0.f, ss = 0.f;
  for (int t = threadIdx.x; t < BB * NN; t += blockDim.x) {
    const int b = t / NN, n = t % NN;
    float v = wout[((size_t)b * CC + o) * NN + n];
    s += v;
    ss += v * v;
  }
  __shared__ float sh[256], sh2[256];
  sh[threadIdx.x] = s;
  sh2[threadIdx.x] = ss;
  __syncthreads();
  for (int st = 128; st > 0; st >>= 1) {
    if (threadIdx.x < st) {
      sh[threadIdx.x] += sh[threadIdx.x + st];
      sh2[threadIdx.x] += sh2[threadIdx.x + st];
    }
    __syncthreads();
  }
  if (threadIdx.x == 0) {
    stats[o] = sh[0];
    stats[CC + o] = sh2[0];
  }
}

// ---------------------------------------------------------------------------
// Kernel 7: BatchNorm (batch stats, biased var) + residual -> out5.
// ---------------------------------------------------------------------------
__global__ void bn_res_kernel(const float* __restrict__ wout,
                              const float* __restrict__ stats,
                              const float* __restrict__ gamma,
                              const float* __restrict__ beta,
                              const float* __restrict__ x,
                              float* __restrict__ out5) {
  const int idx = blockIdx.x * blockDim.x + threadIdx.x;  // over B*C*N
  const int c = (idx / NN) % CC;
  const float invn = 1.f / (float)(BB * NN);
  float mean = stats[c] * invn;
  float var = stats[CC + c] * invn - mean * mean;
  float w = wout[idx];
  out5[idx] = (w - mean) * rsqrtf(var + EPS) * gamma[c] + beta[c] + x[idx];
}

// ---------------------------------------------------------------------------
extern "C" void kernel_launch(void* const* d_in, const int* in_sizes, int n_in,
                              void* d_out, int out_size, void* d_ws,
                              size_t ws_size, hipStream_t stream) {
  const float* x     = (const float*)d_in[0];
  const float* q_w   = (const float*)d_in[1];
  const float* q_b   = (const float*)d_in[2];
  const float* k_w   = (const float*)d_in[3];
  const float* k_b   = (const float*)d_in[4];
  const float* v_w   = (const float*)d_in[5];
  const float* v_b   = (const float*)d_in[6];
  const float* w_w   = (const float*)d_in[7];
  const float* w_b   = (const float*)d_in[8];
  const float* gamma = (const float*)d_in[9];
  const float* beta  = (const float*)d_in[10];

  // Workspace carve-up (floats): ~13.8 MB total.
  float* ws   = (float*)d_ws;
  float* q    = ws;                       // B*CQ*N
  float* k    = q + (size_t)BB * CQ * NN; // B*CQ*N
  float* v    = k + (size_t)BB * CQ * NN; // B*C*N
  float* rmax = v + (size_t)BB * CC * NN; // B*N
  float* rsum = rmax + (size_t)BB * NN;   // B*N
  float* outm = rsum + (size_t)BB * NN;   // B*C*N
  float* wout = outm + (size_t)BB * CC * NN; // B*C*N
  float* stats = wout + (size_t)BB * CC * NN; // 2*C

  float* out5 = (float*)d_out;                       // B*C*N
  float* attn = out5 + (size_t)BB * CC * NN;         // B*N*N

  proj_kernel<<<(BB * NN) / 256, 256, 0, stream>>>(x, q_w, q_b, k_w, k_b, v_w,
                                                   v_b, q, k, v);
  // B*NT = 1024 stripes, 8 waves/block -> 128 blocks
  attn_rowstats_kernel<<<(BB * NT) / 8, 256, 0, stream>>>(q, k, rmax, rsum);
  attn_write_kernel<<<(BB * NT) / 8, 256, 0, stream>>>(q, k, rmax, rsum, attn);
  // B*(C/16)*NT = 4096 tiles, 8 waves/block -> 512 blocks
  out_gemm_kernel<<<(BB * (CC / 16) * NT) / 8, 256, 0, stream>>>(v, attn, outm);
  wproj_kernel<<<(BB * NN) / 256, 256, 0, stream>>>(outm, w_w, w_b, wout);
  stats_kernel<<<CC, 256, 0, stream>>>(wout, stats);
  bn_res_kernel<<<(BB * CC * NN) / 256, 256, 0, stream>>>(wout, stats, gamma,
                                                          beta, x, out5);
}